// cabb_85839216378518
// MI455X (gfx1250) — compile-verified
//
#include <hip/hip_runtime.h>
#include <math.h>

// ---------------------------------------------------------------------------
// MI455X reasoning:
//  * Workload = elementwise transcendental-heavy loss (48 bisections/box,
//    1-2 v_log_f32 per bisection step). Compute-bound on VALU/trans pipe.
//  * 72MB input traffic is negligible at 23.3 TB/s (and fits in 192MB L2).
//  * wave32 ballot early-exit on the bisection loop (reference always runs
//    40 iters; typical convergence is ~10 iters for EPS=0.1).
//  * Deterministic 2-stage reduction; final cross-lane reduce uses
//    V_WMMA_F32_16X16X4_F32 (ones-matrix reduction) -> gfx1250 matrix path.
// ---------------------------------------------------------------------------

#define DEVINL __device__ __forceinline__

constexpr int   N_ITERS    = 40;
constexpr float EPS_BISECT = 0.1f;
constexpr float TINYF      = 1e-12f;
constexpr float E1         = 2.71828182845904523f;   // e^BETA, BETA = 1
constexpr float FOUR_SQRT2 = 5.65685424949238019f;   // 4*sqrt(2)

DEVINL float sl1(float x)  { float a = fabsf(x); return (a < 1.0f) ? 0.5f * a * a : (a - 0.5f); }
DEVINL float sl1p(float x) { return fminf(fmaxf(x, -1.0f), 1.0f); }      // BETA=1
DEVINL float slog(float x) { return __logf(fmaxf(x, TINYF)); }

DEVINL float xi_fn(float w, float lwp, float wh) {
    // xi(w) = smooth_l1((w-wh)/2) + smooth_l1(log(wp)-log(w))
    return sl1((w - wh) * 0.5f) + sl1(lwp - slog(w));
}

template <bool SIGMA>
DEVINL float feval(float w, float lwp, float wh) {
    float x   = (w - wh) * 0.5f;
    float d   = slog(w) - lwp;
    float den = (fabsf(w) > TINYF) ? w : TINYF;
    float xp  = 0.5f * sl1p(x) + sl1p(d) / den;   // xi'
    return SIGMA ? (w * xp) : xp;                 // sigma = w * xi'
}

template <bool SIGMA>
DEVINL float find_min(float u, float v, float lwp, float wh) {
    float fu = feval<SIGMA>(u, lwp, wh);
    float fv = feval<SIGMA>(v, lwp, wh);
    float cu = u, cv = v;
    for (int i = 0; i < N_ITERS; ++i) {
        bool notdone = !((cv - cu) < EPS_BISECT);
        // wave32 vote: once every active lane has converged, stop burning
        // transcendental cycles (reference result is unchanged after 'done').
        if (__builtin_amdgcn_ballot_w32(notdone) == 0u) break;
        float m    = 0.5f * (cu + cv);
        bool  left = feval<SIGMA>(m, lwp, wh) >= 0.0f;
        bool  done = !notdone;
        cu = (done || left)  ? cu : m;
        cv = (done || !left) ? cv : m;
    }
    float m = 0.5f * (cu + cv);
    return (fu >= 0.0f) ? u : ((fv <= 0.0f) ? v : m);
}

__device__ __noinline__ float solve_o1(float wp, float wh, float a1, float b1) {
    float w0  = b1 - a1;
    float lwp = slog(wp);

    bool  condA = fmaxf(w0, wh) < wp;
    float rA    = find_min<false>(fmaxf(w0, wh), wp, lwp, wh);

    bool  condB = fmaxf(w0, wp) < wh;
    float j1u = fmaxf(w0, wp);
    float j1v = fminf(E1 * wp, wh);
    float j2u = fmaxf(fmaxf(w0, 2.0f), fmaxf(wh - 2.0f, E1 * wp));
    float j2v = wh;
    float j3u = fmaxf(w0, fmaxf(wh - 2.0f, E1 * wp));
    float j3v = fminf(E1, wh);
    float disc = sqrtf(fmaxf(1.0f - 32.0f / fmaxf(wh * wh, TINYF), 0.0f));
    float j4v = fminf(fminf(E1 * wp, wh), wh * 0.25f * (1.0f + disc));
    float j5u = fmaxf(j3u, wh * 0.25f * (1.0f - disc));
    float j5v = fminf(E1 * wp, wh);

    float c2 = find_min<false>(j1u, j1v, lwp, wh);
    float c3 = find_min<false>(j2u, j2v, lwp, wh);
    float c4 = find_min<true >(j3u, j3v, lwp, wh);
    float c5 = find_min<true >(j3u, j4v, lwp, wh);
    float c6 = find_min<true >(j5u, j5v, lwp, wh);

    bool small = wh <= FOUR_SQRT2;
    float cands[7] = { w0, wh, c2, c3, c4, c5, c6 };
    bool  valid[7] = { true, true, true, true, small, !small, !small };

    // first-occurrence argmin (matches jnp.argmin tie/NaN-skip behavior of <)
    float best = INFINITY; int bi = 0;
    #pragma unroll
    for (int i = 0; i < 7; ++i) {
        float v = valid[i] ? xi_fn(cands[i], lwp, wh) : INFINITY;
        if (v < best) { best = v; bi = i; }
    }
    float rB = cands[bi];
    return condA ? rA : (condB ? rB : w0);
}

DEVINL void solve_o2(float dp, float wp, float a2, float b2, float* d, float* w) {
    float oh1 = 2.0f * (dp - a2);
    float oh2 = 2.0f * (b2 - dp);
    float w1  = solve_o1(wp, oh1, a2, b2);
    float w2  = solve_o1(wp, oh2, a2, b2);
    float lwp = slog(wp);
    bool  pick1 = xi_fn(w1, lwp, oh1) <= xi_fn(w2, lwp, oh2);
    float dd = pick1 ? (a2 + 0.5f * w1) : (b2 + 0.5f * w2);
    float ww = pick1 ? w1 : w2;
    bool  trivial = wp >= fmaxf(oh1, oh2);
    *d = trivial ? dp : dd;
    *w = trivial ? wp : ww;
}

DEVINL float axis_loss(bool lt, bool rb, float td, float to, float pd, float po,
                       float ppd, float ppo, float cs) {
    float ca01  = 0.5f * (ppd + ppo);
    float da01  = ppo - ppd;
    float a1_01 = td - 0.5f * to;
    float b1_01 = (cs - ca01) / da01;
    float oh01  = 2.0f * (pd - a1_01);
    float w01   = solve_o1(po, oh01, a1_01, b1_01);
    float d01   = a1_01 + 0.5f * w01;

    float ca10  = ca01 + 0.5f;
    float da10  = da01 + 1.0f;
    float a1_10 = -ca10 / da10;
    float b1_10 = td + 0.5f * to;
    float oh10  = 2.0f * (b1_10 - pd);
    float w10   = solve_o1(po, oh10, a1_10, b1_10);
    float d10   = b1_10 - 0.5f * w10;

    float da11 = da01 - 1.0f;
    float a2   = -ca10 / da11;
    float b2   = (cs - ca10) / da11;
    float d11, w11;
    solve_o2(pd, po, a2, b2, &d11, &w11);

    float dl = lt ? (rb ? d11 : d10) : (rb ? d01 : td);
    float wl = lt ? (rb ? w11 : w10) : (rb ? w01 : to);

    float x    = pd - dl;
    float diff = slog(po) - slog(wl);
    return sl1(x) + sl1(diff);
}

// ------------------------- stage 1: per-box loss ---------------------------
__global__ void __launch_bounds__(256)
bbox_loss_kernel(const float* __restrict__ pred,
                 const float* __restrict__ gt,
                 const float* __restrict__ prop,
                 const float* __restrict__ crop,
                 const int*   __restrict__ cases,
                 float*       __restrict__ partials,
                 int N) {
    float lx = 0.0f, ly = 0.0f;
    int stride = gridDim.x * blockDim.x;
    for (int i = blockIdx.x * blockDim.x + threadIdx.x; i < N; i += stride) {
        float4 p  = reinterpret_cast<const float4*>(prop)[i];
        float4 g  = reinterpret_cast<const float4*>(gt)[i];
        float4 pr = reinterpret_cast<const float4*>(pred)[i];
        float2 cs = reinterpret_cast<const float2*>(crop)[i];
        int4   c  = reinterpret_cast<const int4*>(cases)[i];

        // bbox2delta
        float pw = p.z - p.x + 1.0f, ph = p.w - p.y + 1.0f;
        float px = (p.x + p.z) * 0.5f, py = (p.y + p.w) * 0.5f;
        float gw = g.z - g.x + 1.0f, gh = g.w - g.y + 1.0f;
        float gx = (g.x + g.z) * 0.5f, gy = (g.y + g.w) * 0.5f;
        float dx = (gx - px) / pw, dy = (gy - py) / ph;
        float t2 = gw / pw, t3 = gh / ph;            // exp(log(gw/pw))

        float pc0 = pr.x, pc1 = pr.y;
        float pc2 = __expf(pr.z), pc3 = __expf(pr.w);

        lx += axis_loss(c.x != 0, c.y != 0, dx, t2, pc0, pc2, p.x, p.z, cs.x);
        ly += axis_loss(c.z != 0, c.w != 0, dy, t3, pc1, pc3, p.y, p.w, cs.y);
    }

    // wave32 tree reduce
    #pragma unroll
    for (int off = 16; off; off >>= 1) {
        lx += __shfl_xor(lx, off, 32);
        ly += __shfl_xor(ly, off, 32);
    }
    __shared__ float sx[8], sy[8];
    int lane = threadIdx.x & 31, wid = threadIdx.x >> 5;
    if (lane == 0) { sx[wid] = lx; sy[wid] = ly; }
    __syncthreads();
    if (threadIdx.x == 0) {
        float ax = 0.0f, ay = 0.0f;
        #pragma unroll
        for (int w = 0; w < 8; ++w) { ax += sx[w]; ay += sy[w]; }
        partials[2 * blockIdx.x]     = ax;
        partials[2 * blockIdx.x + 1] = ay;
    }
}

// ----------------- stage 2: single-wave WMMA ones-reduction ----------------
typedef __attribute__((ext_vector_type(2))) float v2f;
typedef __attribute__((ext_vector_type(8))) float v8f;

__global__ void __launch_bounds__(32)
wmma_reduce_kernel(const float* __restrict__ partials, int nb,
                   float* __restrict__ out, float invN) {
    int lane = threadIdx.x;                 // exactly 32 lanes, EXEC all 1s
    float sx = 0.0f, sy = 0.0f;
    for (int i = lane; i < nb; i += 32) {
        sx += partials[2 * i];
        sy += partials[2 * i + 1];
    }
    // A (16x4 f32): lane L<16 holds (M=L,K=0),(M=L,K=1); lane L>=16 holds K=2,3.
    // Putting (sx,sy) in every lane makes column K∈{0,2} carry x-partials and
    // K∈{1,3} carry y-partials.
    v2f a; a.x = sx; a.y = sy;
    // B (4x16): VGPR0 = rows K=0 (lanes 0-15) / K=2 (lanes 16-31), VGPR1 = K=1/K=3.
    // Selector: columns 0-7 sum the x rows (K=0,2), columns 8-15 the y rows (K=1,3).
    int n = lane & 15;
    v2f b; b.x = (n < 8) ? 1.0f : 0.0f;
           b.y = (n < 8) ? 0.0f : 1.0f;
    v8f c = {};
    c = __builtin_amdgcn_wmma_f32_16x16x4_f32(false, a, false, b, (short)0, c,
                                              false, false);
    // D[m][n<8] = sx[m]+sx[m+16]; D[m][n>=8] = sy[m]+sy[m+16].
    // Sum this lane's 8 D rows (M=0..7 for lanes<16, M=8..15 otherwise):
    float s = c[0] + c[1] + c[2] + c[3] + c[4] + c[5] + c[6] + c[7];
    s += __shfl_xor(s, 16, 32);   // fold N-column halves: lane0 -> Σx, lane8 -> Σy
    if (lane == 0) out[0] = s * invN;
    if (lane == 8) out[1] = s * invN;
}

// ---------------------------------------------------------------------------
extern "C" void kernel_launch(void* const* d_in, const int* in_sizes, int n_in,
                              void* d_out, int out_size, void* d_ws, size_t ws_size,
                              hipStream_t stream) {
    const float* pred  = (const float*)d_in[0];
    const float* gt    = (const float*)d_in[1];
    const float* prop  = (const float*)d_in[2];
    const float* crop  = (const float*)d_in[3];
    const int*   cases = (const int*)d_in[4];
    float*       out   = (float*)d_out;

    int N = in_sizes[0] / 4;                       // pred is (N,4)
    const int THREADS = 256;
    int blocks = (N + THREADS - 1) / THREADS;
    if (blocks > 4096) blocks = 4096;
    size_t ws_cap = ws_size / (2 * sizeof(float));
    if ((size_t)blocks > ws_cap) blocks = (int)ws_cap;
    if (blocks < 1) blocks = 1;

    float* partials = (float*)d_ws;

    bbox_loss_kernel<<<blocks, THREADS, 0, stream>>>(pred, gt, prop, crop, cases,
                                                     partials, N);
    wmma_reduce_kernel<<<1, 32, 0, stream>>>(partials, blocks, out,
                                             1.0f / (float)N);
}